// DiffGSTileSampler_57552561767185
// MI455X (gfx1250) — compile-verified
//
#include <hip/hip_runtime.h>

typedef __attribute__((ext_vector_type(2))) float v2f;
typedef __attribute__((ext_vector_type(8))) float v8f;

#define NP 1024
#define IMH 256
#define IMW 256
#define NPIX (IMH * IMW)

// ---------------------------------------------------------------------------
// Kernel 1: depth-sort (stable O(P^2) rank sort, P=1024 -> trivial) and
// expand each gaussian's inverse covariance into the 6 coefficients of the
// quadratic form  q(x,y) = c0*x^2 + c1*xy + c2*y^2 + c3*x + c4*y + c5,
// padded to 8 floats so it maps directly onto the K=8 WMMA A-operand.
// Also writes depth-sorted colors (padded to float4).
// ---------------------------------------------------------------------------
__global__ __launch_bounds__(NP) void gs_preprocess(
    const float* __restrict__ means,   // P x 2
    const float* __restrict__ covs,    // P x 3 (a, b, c)
    const float* __restrict__ depth,   // P
    const float* __restrict__ colors,  // P x 3
    float* __restrict__ coef8,         // out: P x 8, depth-sorted
    float* __restrict__ col4)          // out: P x 4, depth-sorted
{
    __shared__ float sdepth[NP];
    const int i = threadIdx.x;
    const float di = depth[i];
    sdepth[i] = di;
    __syncthreads();

    // stable rank: position of gaussian i in ascending-depth order
    int rank = 0;
    for (int j = 0; j < NP; ++j) {
        const float dj = sdepth[j];
        rank += (int)((dj < di) | ((dj == di) & (j < i)));
    }

    const float mx = means[2 * i + 0];
    const float my = means[2 * i + 1];
    const float a = covs[3 * i + 0];
    const float b = covs[3 * i + 1];
    const float c = covs[3 * i + 2];
    const float det = a * c - b * b;
    const float ia =  c / det;
    const float ib = -b / det;
    const float ic =  a / det;

    // q = ia*(x-mx)^2 + 2 ib (x-mx)(y-my) + ic*(y-my)^2 expanded:
    float* o = coef8 + rank * 8;
    o[0] = ia;                              // x^2
    o[1] = 2.0f * ib;                       // x*y
    o[2] = ic;                              // y^2
    o[3] = -2.0f * (ia * mx + ib * my);     // x
    o[4] = -2.0f * (ic * my + ib * mx);     // y
    o[5] = ia * mx * mx + 2.0f * ib * mx * my + ic * my * my;  // 1
    o[6] = 0.0f;                            // K padding
    o[7] = 0.0f;

    float* oc = col4 + rank * 4;
    oc[0] = colors[3 * i + 0];
    oc[1] = colors[3 * i + 1];
    oc[2] = colors[3 * i + 2];
    oc[3] = 0.0f;
}

// ---------------------------------------------------------------------------
// Kernel 2: render. 256 threads = 8 waves; each wave owns 16 pixels (one
// pixel per lane-pair n / n+16). Per 16-gaussian tile:
//   q_tile(16 gaussians x 16 pixels) = A(16x8) * B(8x16)   via two
//   V_WMMA_F32_16X16X4_F32 accumulating ops. B (pixel features) is
//   loop-invariant; only A is reloaded from LDS each tile.
// The D-matrix layout hands lane n gaussians M=0..7 and lane n+16 gaussians
// M=8..15 of the same pixel, so each lane serially composites its 8
// gaussians locally, and the halves are merged with the associative rule
//   S = S_lo + t_lo*S_hi ; t = t_lo*t_hi    (via __shfl_xor 16).
// ---------------------------------------------------------------------------
__global__ __launch_bounds__(256) void gs_render(
    const float* __restrict__ coef8,
    const float* __restrict__ col4,
    float* __restrict__ out)
{
    __shared__ float scoef[NP * 8];  // 32 KB
    __shared__ float scol[NP * 4];   // 16 KB

    for (int idx = threadIdx.x; idx < NP * 8; idx += 256) scoef[idx] = coef8[idx];
    for (int idx = threadIdx.x; idx < NP * 4; idx += 256) scol[idx]  = col4[idx];
    __syncthreads();

    const int lane = threadIdx.x & 31;
    const int wave = threadIdx.x >> 5;
    const int sub  = lane & 15;   // pixel / gaussian-row index within tile
    const int hi   = lane >> 4;   // 0: K=0,1 & gaussians 0-7 ; 1: K=2,3 & 8-15

    const int pix = blockIdx.x * 128 + wave * 16 + sub;
    const int row = pix >> 8;
    const int col = pix & 255;
    const float x = (float)col + 0.5f;
    const float y = (float)row + 0.5f;

    // B operand, feature order K = [x^2, xy, y^2, x, y, 1, 0, 0].
    // B layout (4x16 f32): lanes 0-15 = pixel n, VGPR0/1 = K0/K1;
    //                      lanes 16-31 = pixel n, VGPR0/1 = K2/K3.
    v2f B0, B1;
    if (hi == 0) { B0.x = x * x; B0.y = x * y; B1.x = y;    B1.y = 1.0f; }
    else         { B0.x = y * y; B0.y = x;     B1.x = 0.0f; B1.y = 0.0f; }

    float T  = 1.0f;                 // running transmittance (replicated n / n+16)
    float Cr = 0.0f, Cg = 0.0f, Cb = 0.0f;

    for (int g0 = 0; g0 < NP; g0 += 16) {
        // A operand: lane holds gaussian row M=sub, K-pair selected by `hi`.
        const float* cf = &scoef[(g0 + sub) * 8 + hi * 2];
        v2f A0; A0.x = cf[0]; A0.y = cf[1];
        v2f A1; A1.x = cf[4]; A1.y = cf[5];   // K=4..7 block (zeros for hi half pad)

        v8f q = {0.f, 0.f, 0.f, 0.f, 0.f, 0.f, 0.f, 0.f};
        q = __builtin_amdgcn_wmma_f32_16x16x4_f32(false, A0, false, B0,
                                                  (short)0, q, false, false);
        q = __builtin_amdgcn_wmma_f32_16x16x4_f32(false, A1, false, B1,
                                                  (short)0, q, false, false);

        // local composite of this lane's 8 gaussians (t starts at 1)
        float t = 1.0f;
        float Sr = 0.0f, Sg = 0.0f, Sb = 0.0f;
        const int gbase = g0 + hi * 8;
#pragma unroll
        for (int m = 0; m < 8; ++m) {
            float alpha = fminf(0.99f, __expf(-0.5f * q[m]));
            alpha = (alpha < (1.0f / 255.0f)) ? 0.0f : alpha;
            const float w = alpha * t;
            const float* cc = &scol[(gbase + m) * 4];
            Sr += w * cc[0];
            Sg += w * cc[1];
            Sb += w * cc[2];
            t *= (1.0f - alpha);
        }

        // merge the two 8-gaussian halves of this pixel (lane n <-> n+16)
        const float pt  = __shfl_xor(t,  16, 32);
        const float pSr = __shfl_xor(Sr, 16, 32);
        const float pSg = __shfl_xor(Sg, 16, 32);
        const float pSb = __shfl_xor(Sb, 16, 32);

        const float cSr = hi ? (pSr + pt * Sr) : (Sr + t * pSr);
        const float cSg = hi ? (pSg + pt * Sg) : (Sg + t * pSg);
        const float cSb = hi ? (pSb + pt * Sb) : (Sb + t * pSb);
        const float cT  = t * pt;

        Cr += T * cSr;
        Cg += T * cSg;
        Cb += T * cSb;
        T  *= cT;
    }

    if (hi == 0) {
        out[0 * NPIX + pix] = Cr;
        out[1 * NPIX + pix] = Cg;
        out[2 * NPIX + pix] = Cb;
    }
}

extern "C" void kernel_launch(void* const* d_in, const int* in_sizes, int n_in,
                              void* d_out, int out_size, void* d_ws, size_t ws_size,
                              hipStream_t stream) {
    const float* means  = (const float*)d_in[0];  // P x 2
    const float* covs   = (const float*)d_in[1];  // P x 3
    const float* depth  = (const float*)d_in[2];  // P
    const float* colors = (const float*)d_in[3];  // P x 3
    // d_in[4], d_in[5] are height/width scalars; fixed at 256 here.

    float* coef8 = (float*)d_ws;            // P*8 floats = 32 KB
    float* col4  = (float*)d_ws + NP * 8;   // P*4 floats = 16 KB

    gs_preprocess<<<1, NP, 0, stream>>>(means, covs, depth, colors, coef8, col4);
    gs_render<<<NPIX / 128, 256, 0, stream>>>(coef8, col4, (float*)d_out);
}